// IdentityKeyValuePoolingAttention_63977832841885
// MI455X (gfx1250) — compile-verified
//
#include <hip/hip_runtime.h>

// IdentityKeyValuePoolingAttention for MI455X (gfx1250, wave32).
//
// Shapes: V (B=16, L=32768, D=512) fp32, Q (1,1,512) fp32, out (16,512) fp32.
// Bandwidth-bound: V = 1 GiB, floor ~46us @ 23.3 TB/s. Single-pass online
// softmax so V is streamed exactly once. Scores via V_WMMA_F32_16X16X4_F32
// (A = query replicated over M rows -> D rows identical -> lane L holds
// score(row = L%16) in c[0]).
//
// Phase 1: grid (CHUNKS, B), 8 waves/block, wave w = head w. Per wave:
//   online softmax over 512 rows in groups of 16 -> partial (m, s, o[64]).
// Phase 2: combine 64 chunk-partials per (b,h), normalize, write output.
//
// Round-1 change: literal-index probability broadcasts use v_readlane_b32
// (SGPR result, consumed directly by v_fmac) instead of ds_bpermute, cutting
// 16 LDS ops + dscnt waits per 16-row group out of the inner loop.

typedef __attribute__((ext_vector_type(2))) float v2f;
typedef __attribute__((ext_vector_type(8))) float v8f;

#define B_      16
#define L_      32768
#define D_      512
#define H_      8
#define DS_     64
#define CHUNKS_ 64
#define ROWS_   (L_ / CHUNKS_)       // 512 rows per (block, head)
#define GROUPS_ (ROWS_ / 16)         // 32 groups of 16 rows
#define NP_     (B_ * CHUNKS_ * H_)  // 8192 partial records

__device__ __forceinline__ float lane_bcast(float v, int srcLane) {
    // srcLane is a compile-time literal at every call site -> v_readlane_b32
    return __uint_as_float(__builtin_amdgcn_readlane(__float_as_uint(v), srcLane));
}

__global__ __launch_bounds__(256) void pool_attn_phase1(
    const float* __restrict__ V, const float* __restrict__ Q,
    float* __restrict__ mbuf, float* __restrict__ sbuf,
    float* __restrict__ obuf)
{
    const int chunk = blockIdx.x;
    const int b     = blockIdx.y;
    const int wave  = threadIdx.x >> 5;
    const int lane  = threadIdx.x & 31;
    const int h     = wave;          // one wave per head
    const int half  = lane >> 4;     // K-half selector for WMMA A/B layout
    const int rsub  = lane & 15;     // row-in-group this lane carries

    const float scale = 0.044194173824159216f; // 1/sqrt(512), folded into Q

    // A operand: 16x4 f32, A[m][k] = q_scaled[4k + k'] replicated over m.
    // Lanes 0-15 hold K={4k,4k+1}, lanes 16-31 hold K={4k+2,4k+3}.
    v2f a[16];
    {
        const float* qh = Q + h * DS_ + 2 * half;
        #pragma unroll
        for (int k = 0; k < 16; ++k) {
            a[k].x = qh[4 * k + 0] * scale;
            a[k].y = qh[4 * k + 1] * scale;
        }
    }

    float m_run = -3.0e38f;
    float s_run = 0.0f;
    float ox = 0.0f, oy = 0.0f;   // lane owns output dims {2*lane, 2*lane+1}

    const size_t vbase = (size_t)b * L_ * D_ + (size_t)h * DS_;
    const int l0c = chunk * ROWS_;

    for (int g = 0; g < GROUPS_; ++g) {
        const int l0 = l0c + g * 16;

        // B operand: 4x16 f32, B[k][n] = V[row n][4k + k'].
        // Lane L reads row (L%16); lanes 0-15 K={4k,4k+1}, 16-31 K={4k+2,4k+3}.
        const float* vb = V + vbase + (size_t)(l0 + rsub) * D_ + 2 * half;

        v8f c0 = {};
        v8f c1 = {};
        #pragma unroll
        for (int k = 0; k < 16; k += 2) {
            v2f b0, b1;
            b0.x = vb[4 * k + 0];
            b0.y = vb[4 * k + 1];
            b1.x = vb[4 * (k + 1) + 0];
            b1.y = vb[4 * (k + 1) + 1];
            c0 = __builtin_amdgcn_wmma_f32_16x16x4_f32(
                     false, a[k], false, b0, (short)0, c0, false, false);
            c1 = __builtin_amdgcn_wmma_f32_16x16x4_f32(
                     false, a[k + 1], false, b1, (short)0, c1, false, false);
        }
        // D rows are identical (A replicated) -> c[0] = score(row rsub), scaled.
        float s = c0[0] + c1[0];

        // Group max over the 16 distinct rows (both halves duplicate values).
        float gm = s;
        gm = fmaxf(gm, __shfl_xor(gm, 1, 32));
        gm = fmaxf(gm, __shfl_xor(gm, 2, 32));
        gm = fmaxf(gm, __shfl_xor(gm, 4, 32));
        gm = fmaxf(gm, __shfl_xor(gm, 8, 32));

        const float m_new = fmaxf(m_run, gm);
        const float alpha = __expf(m_run - m_new);
        m_run = m_new;

        const float p = __expf(s - m_new);
        float ps = p;
        ps += __shfl_xor(ps, 1, 32);
        ps += __shfl_xor(ps, 2, 32);
        ps += __shfl_xor(ps, 4, 32);
        ps += __shfl_xor(ps, 8, 32);
        s_run = s_run * alpha + ps;

        ox *= alpha;
        oy *= alpha;

        // Accumulate o[d] += p_r * V[row r][d]; tile is L0-resident (just read).
        // p_r broadcast via v_readlane (lane r holds row r's p), SGPR operand
        // feeds the FMAs directly.
        const float* vr = V + vbase + (size_t)l0 * D_ + 2 * lane;
        #pragma unroll
        for (int r = 0; r < 16; ++r) {
            const float pr = lane_bcast(p, r);
            ox = fmaf(pr, vr[(size_t)r * D_ + 0], ox);
            oy = fmaf(pr, vr[(size_t)r * D_ + 1], oy);
        }
    }

    const int idx = (b * CHUNKS_ + chunk) * H_ + h;
    if (lane == 0) {
        mbuf[idx] = m_run;
        sbuf[idx] = s_run;
    }
    float* op = obuf + (size_t)idx * DS_ + 2 * lane;
    op[0] = ox;
    op[1] = oy;
}

__global__ __launch_bounds__(256) void pool_attn_phase2(
    const float* __restrict__ mbuf, const float* __restrict__ sbuf,
    const float* __restrict__ obuf, float* __restrict__ out)
{
    const int b    = blockIdx.x;
    const int wave = threadIdx.x >> 5;
    const int lane = threadIdx.x & 31;
    const int h    = wave;

    const int i0 = (b * CHUNKS_ + lane) * H_ + h;
    const int i1 = (b * CHUNKS_ + lane + 32) * H_ + h;
    const float m0 = mbuf[i0];
    const float m1 = mbuf[i1];

    float M = fmaxf(m0, m1);
    M = fmaxf(M, __shfl_xor(M, 1, 32));
    M = fmaxf(M, __shfl_xor(M, 2, 32));
    M = fmaxf(M, __shfl_xor(M, 4, 32));
    M = fmaxf(M, __shfl_xor(M, 8, 32));
    M = fmaxf(M, __shfl_xor(M, 16, 32));

    float S = sbuf[i0] * __expf(m0 - M) + sbuf[i1] * __expf(m1 - M);
    S += __shfl_xor(S, 1, 32);
    S += __shfl_xor(S, 2, 32);
    S += __shfl_xor(S, 4, 32);
    S += __shfl_xor(S, 8, 32);
    S += __shfl_xor(S, 16, 32);

    float ox = 0.0f, oy = 0.0f;
    for (int c = 0; c < CHUNKS_; ++c) {
        const int idx = (b * CHUNKS_ + c) * H_ + h;
        const float w = __expf(mbuf[idx] - M);
        const float* op = obuf + (size_t)idx * DS_ + 2 * lane;
        ox = fmaf(w, op[0], ox);
        oy = fmaf(w, op[1], oy);
    }

    const float inv = 1.0f / S;
    float* o = out + (size_t)b * D_ + h * DS_ + 2 * lane;
    o[0] = ox * inv;
    o[1] = oy * inv;
}

extern "C" void kernel_launch(void* const* d_in, const int* in_sizes, int n_in,
                              void* d_out, int out_size, void* d_ws, size_t ws_size,
                              hipStream_t stream)
{
    (void)in_sizes; (void)n_in; (void)out_size; (void)ws_size;

    const float* V = (const float*)d_in[0];  // (16, 32768, 512) fp32
    const float* Q = (const float*)d_in[1];  // (1, 1, 512) fp32
    float* out = (float*)d_out;              // (16, 512) fp32

    // Workspace layout: m[NP_], s[NP_], o[NP_*64]  (~2.1 MB total)
    float* mbuf = (float*)d_ws;
    float* sbuf = mbuf + NP_;
    float* obuf = sbuf + NP_;

    dim3 grid1(CHUNKS_, B_);
    pool_attn_phase1<<<grid1, 256, 0, stream>>>(V, Q, mbuf, sbuf, obuf);
    pool_attn_phase2<<<B_, 256, 0, stream>>>(mbuf, sbuf, obuf, out);
}